// Energy_layer_83107617178053
// MI455X (gfx1250) — compile-verified
//
#include <hip/hip_runtime.h>
#include <cmath>

// ---------------------------------------------------------------------------
// Problem constants (match reference)
// ---------------------------------------------------------------------------
constexpr int NXg = 4096;
constexpr int NYg = 2048;
constexpr float FLUXC  = 300.0f * (6.0f / 4096.0f);   // 0.439453125
constexpr float SQRT2C = 1.41421f;

// Tiling: 256 threads per block, 64x16 tile.
// LDS halo tile: 18 rows x 66 logical cols, padded to row stride 72 floats so
// the 64 interior columns (LDS col 4..67) are 16-byte aligned for b128 ops.
// Col 3 = west halo, col 68 = east halo.
constexpr int TX = 64;
constexpr int TY = 16;
constexpr int HX = TX + 2;          // 66 logical halo cols
constexpr int HY = TY + 2;          // 18 halo rows
constexpr int ROWS = 72;            // padded LDS row stride (floats)
constexpr int NTHREADS = 256;

typedef float v2f __attribute__((ext_vector_type(2)));
typedef float v4f __attribute__((ext_vector_type(4)));
typedef float v8f __attribute__((ext_vector_type(8)));
typedef __attribute__((__vector_size__(4 * sizeof(int)))) int vi4;

// ---------------------------------------------------------------------------
// CDNA5 async global->LDS staging (ASYNCcnt path).
// Probe-verified signatures:
//   b32 : (AS1 int*  global_src, AS3 int*  lds_dst, Ii offset, Ii cpol)
//   b128: (AS1 vi4*  global_src, AS3 vi4*  lds_dst, Ii offset, Ii cpol)
// ---------------------------------------------------------------------------
#define AS1 __attribute__((address_space(1)))
#define AS3 __attribute__((address_space(3)))

#if __has_builtin(__builtin_amdgcn_global_load_async_to_lds_b32)
#define ASYNC_LOAD_B32(gsrc, ldst)                                          \
  __builtin_amdgcn_global_load_async_to_lds_b32(                            \
      (AS1 int*)(void*)const_cast<float*>(gsrc),                            \
      (AS3 int*)(void*)(ldst), 0, 0)
#else
__device__ __forceinline__ void async_load_b32_asm(const void* g, void* l) {
  unsigned loff = (unsigned)(unsigned long long)(AS3 void*)l;
  unsigned long long ga = (unsigned long long)g;
  asm volatile("global_load_async_to_lds_b32 %0, %1, off"
               :: "v"(loff), "v"(ga) : "memory");
}
#define ASYNC_LOAD_B32(gsrc, ldst) async_load_b32_asm((const void*)(gsrc), (void*)(ldst))
#endif

#if __has_builtin(__builtin_amdgcn_global_load_async_to_lds_b128)
#define ASYNC_LOAD_B128(gsrc, ldst)                                         \
  __builtin_amdgcn_global_load_async_to_lds_b128(                           \
      (AS1 vi4*)(void*)const_cast<float*>(gsrc),                            \
      (AS3 vi4*)(void*)(ldst), 0, 0)
#else
#define ASYNC_LOAD_B128(gsrc, ldst) do {                                    \
  ASYNC_LOAD_B32((gsrc) + 0, (float*)(ldst) + 0);                           \
  ASYNC_LOAD_B32((gsrc) + 1, (float*)(ldst) + 1);                           \
  ASYNC_LOAD_B32((gsrc) + 2, (float*)(ldst) + 2);                           \
  ASYNC_LOAD_B32((gsrc) + 3, (float*)(ldst) + 3);                           \
} while (0)
#endif

#if __has_builtin(__builtin_amdgcn_s_wait_asynccnt)
#define WAIT_ASYNC0() __builtin_amdgcn_s_wait_asynccnt(0)
#else
#define WAIT_ASYNC0() asm volatile("s_wait_asynccnt 0" ::: "memory")
#endif

// Sequential boundary overrides, exactly in reference order.
__device__ __forceinline__ float b_effective(float v, int r, int c) {
  if (r == 1 && c >= 1)        v = 0.0f;
  if (r == NYg - 2 && c >= 1)  v = 0.0f;
  if (c == 1)                  v = 0.0f;
  if (c == NXg - 1)            v = 3.0f;
  if (r == 0)                  v = 3.0f;
  if (r == NYg - 1)            v = 3.0f;
  return v;
}

__device__ __forceinline__ int reflect_row(int r) {
  return (r < 0) ? 1 : ((r >= NYg) ? (NYg - 2) : r);
}
__device__ __forceinline__ int reflect_col(int c) {
  return (c < 0) ? 1 : ((c >= NXg) ? (NXg - 2) : c);
}

// ---------------------------------------------------------------------------
// Kernel 1: tiled stencil + pointwise loss + per-block partial reduction.
// ---------------------------------------------------------------------------
__global__ __launch_bounds__(NTHREADS)
void energy_stencil_kernel(const float* __restrict__ layout,
                           const float* __restrict__ heat,
                           const float* __restrict__ flow,
                           float* __restrict__ out_hbc,
                           float* __restrict__ out_eqm,
                           float* __restrict__ partials)
{
  __shared__ float heatT[HY * ROWS];   // staged heat -> heat_bc in place
  __shared__ float bndT[HY * ROWS];    // staged boundary -> effective b
  __shared__ float redA[NTHREADS];
  __shared__ float redB[NTHREADS];

  const int tid = threadIdx.x;
  const int x0 = blockIdx.x * TX - 1;   // global col of logical halo col 0
  const int y0 = blockIdx.y * TY - 1;   // global row of halo row 0
  const size_t planeN = (size_t)NXg * NYg;
  const float* bnd = layout + planeN;   // layout[:,1,:,:] = boundary channel
  const float* uflow = flow;            // flow[:,0]
  const float* vflow = flow + planeN;   // flow[:,1]

  // ---- Phase 1a: async-stage interior columns as b128 (aligned both sides)
  // Tasks: 2 arrays * 18 rows * 16 four-float chunks = 576.
  constexpr int CH = TX / 4;                      // 16 chunks per row
  for (int t = tid; t < 2 * HY * CH; t += NTHREADS) {
    const int a = t / (HY * CH);                  // 0 = heat, 1 = bnd
    const int q = t - a * (HY * CH);
    const int r = q / CH;                         // halo row 0..17
    const int j = q - r * CH;                     // chunk 0..15
    const int gr = reflect_row(y0 + r);
    const size_t goff = (size_t)gr * NXg + (size_t)blockIdx.x * TX + 4 * j;
    const int lidx = r * ROWS + 4 + 4 * j;        // 16B-aligned LDS dest
    if (a == 0) ASYNC_LOAD_B128(heat + goff, &heatT[lidx]);
    else        ASYNC_LOAD_B128(bnd + goff, &bndT[lidx]);
  }
  // ---- Phase 1b: async-stage reflected edge columns as b32 (72 tasks) ----
  if (tid < 2 * HY * 2) {
    const int a = tid / (HY * 2);
    const int e = tid - a * (HY * 2);
    const int r = e >> 1;
    const int side = e & 1;                       // 0 = west, 1 = east
    const int gr = reflect_row(y0 + r);
    const int gc = reflect_col(side ? (x0 + HX - 1) : x0);
    const size_t goff = (size_t)gr * NXg + gc;
    const int lidx = r * ROWS + (side ? (4 + TX) : 3);
    if (a == 0) ASYNC_LOAD_B32(heat + goff, &heatT[lidx]);
    else        ASYNC_LOAD_B32(bnd + goff, &bndT[lidx]);
  }
  WAIT_ASYNC0();
  __syncthreads();

  // ---- Phase 2: apply BC overrides once per staged cell, in place ---------
  for (int i = tid; i < HY * HX; i += NTHREADS) {
    const int r = i / HX;
    const int q = i - r * HX;                     // logical col 0..65
    const int lidx = r * ROWS + 3 + q;
    const int gr = reflect_row(y0 + r);
    const int gc = reflect_col(x0 + q);
    const float be = b_effective(bndT[lidx], gr, gc);
    bndT[lidx] = be;                               // effective boundary
    heatT[lidx] = (be == 1.0f) ? 0.0f : heatT[lidx];  // heat_bc
  }
  __syncthreads();

  // ---- Phase 3: stencil; one thread = 4 consecutive cols of one row -------
  const int row   = tid >> 4;                     // 0..15
  const int chunk = tid & 15;                     // 0..15
  const int hr    = row + 1;
  const int base  = hr * ROWS + 4 + 4 * chunk;    // 16B-aligned

  const v4f c4 = *(const v4f*)&heatT[base];
  const v4f n4 = *(const v4f*)&heatT[base - ROWS];
  const v4f s4 = *(const v4f*)&heatT[base + ROWS];
  const v4f b4 = *(const v4f*)&bndT[base];
  const float wS = heatT[base - 1];
  const float eS = heatT[base + 4];
  v4f w4, e4;
  w4[0] = wS;    w4[1] = c4[0]; w4[2] = c4[1]; w4[3] = c4[2];
  e4[0] = c4[1]; e4[1] = c4[2]; e4[2] = c4[3]; e4[3] = eS;

  const int gr = blockIdx.y * TY + row;
  const size_t off = (size_t)gr * NXg + (size_t)blockIdx.x * TX + 4 * chunk;
  const v4f u4 = *(const v4f*)(uflow + off);      // aligned b128
  const v4f v4 = *(const v4f*)(vflow + off);      // aligned b128

  float sle2 = 0.0f, slf2 = 0.0f;
  v4f eqm4;
  #pragma unroll
  for (int e = 0; e < 4; ++e) {
    const float c  = c4[e];
    const float E  = e4[e];
    const float W  = w4[e];
    const float Nv = n4[e];
    const float Sv = s4[e];
    const float b  = b4[e];
    const float u  = u4[e];
    const float v  = v4[e];

    const float dx  = 0.5f * (E - W);
    const float dy  = 0.5f * (Sv - Nv);
    const float fdx = E - c;
    const float fdy = Nv - c;
    const float bdx = c - W;
    const float bdy = c - Sv;
    const float adv = u * dx + v * dy;

    const int bi = (int)b;                        // b in {0..11}, exact
    const float em = (bi != 1 && bi != 2) ? 1.0f : 0.0f;
    float le = adv * em;
    float lf = 0.0f;
    switch (bi) {
      case 4:  le += fdx;            lf = fabsf(fdx)       - FLUXC;          break;
      case 5:  le += fdy;            lf = fabsf(fdy)       - FLUXC;          break;
      case 6:  le += bdx;            lf = fabsf(bdx)       - FLUXC;          break;
      case 7:  le += bdy;            lf = fabsf(bdy)       - FLUXC;          break;
      case 8:  le += fdx; le += fdy; lf = fabsf(fdx + fdy) - SQRT2C * FLUXC; break;
      case 9:  le += fdy; le += bdx; lf = fabsf(fdy + bdx) - SQRT2C * FLUXC; break;
      case 10: le += bdx; le += bdy; lf = fabsf(bdx + bdy) - SQRT2C * FLUXC; break;
      case 11: le += fdx; le += bdy; lf = fabsf(fdx + bdy) - SQRT2C * FLUXC; break;
      default: break;
    }
    sle2 += le * le;
    slf2 += lf * lf;
    eqm4[e] = b * em;
  }

  // Outputs: d_out+1 base breaks 16B alignment, so scalar (coalesced) stores.
  #pragma unroll
  for (int e = 0; e < 4; ++e) {
    out_hbc[off + e] = c4[e];                     // heat_bc (masked in LDS)
    out_eqm[off + e] = eqm4[e];                   // eq_mask
  }

  // ---- Phase 4: deterministic block tree reduction ------------------------
  redA[tid] = sle2;
  redB[tid] = slf2;
  __syncthreads();
  for (int s = NTHREADS / 2; s > 0; s >>= 1) {
    if (tid < s) {
      redA[tid] += redA[tid + s];
      redB[tid] += redB[tid + s];
    }
    __syncthreads();
  }
  if (tid == 0) {
    const int bid = blockIdx.y * gridDim.x + blockIdx.x;
    partials[2 * bid + 0] = redA[0];
    partials[2 * bid + 1] = redB[0];
  }
}

// ---------------------------------------------------------------------------
// Kernel 2: deterministic final reduction; last 32->1 step uses one
// v_wmma_f32_16x16x4_f32 (A = per-lane (Se,Sf), B = 0/1 selector matrix so
// column 0 of D collects Se and column 1 collects Sf).
// ---------------------------------------------------------------------------
__global__ __launch_bounds__(NTHREADS)
void energy_reduce_kernel(const float* __restrict__ partials, int nblocks,
                          float* __restrict__ out0)
{
  __shared__ float rA[NTHREADS];
  __shared__ float rB[NTHREADS];
  __shared__ float sred[32];

  const int tid = threadIdx.x;
  float a = 0.0f, b = 0.0f;
  for (int i = tid; i < nblocks; i += NTHREADS) {
    a += partials[2 * i + 0];
    b += partials[2 * i + 1];
  }
  rA[tid] = a;
  rB[tid] = b;
  __syncthreads();
  for (int s = NTHREADS / 2; s >= 32; s >>= 1) {
    if (tid < s) {
      rA[tid] += rA[tid + s];
      rB[tid] += rB[tid + s];
    }
    __syncthreads();
  }

  if (tid < 32) {   // wave 0 only: EXEC all-ones inside, as WMMA requires
    // A(16x4) per ISA layout: lanes 0-15 hold (M=lane, K=0/1) in VGPR0/1,
    // lanes 16-31 hold (M=lane-16, K=2/3).
    v2f A;
    A.x = rA[tid];
    A.y = rB[tid];
    // B(4x16): B[k][0]=1 for k in {0,2} (energy), B[k][1]=1 for k in {1,3}.
    v2f B;
    B.x = (tid == 0 || tid == 16) ? 1.0f : 0.0f;
    B.y = (tid == 1 || tid == 17) ? 1.0f : 0.0f;
    v8f C = {};
    v8f D = __builtin_amdgcn_wmma_f32_16x16x4_f32(
        false, A, false, B, (short)0, C, false, false);
    // Each lane sums its 8 C/D rows; column 0 lives in lanes {0,16},
    // column 1 in lanes {1,17}.
    float s = D[0] + D[1] + D[2] + D[3] + D[4] + D[5] + D[6] + D[7];
    sred[tid] = s;
    if (tid == 0) {
      constexpr float INV_N = 1.0f / (float)((long)NXg * NYg);  // 2^-23, exact
      const float mseE = (sred[0] + sred[16]) * INV_N;
      const float mseF = (sred[1] + sred[17]) * INV_N;
      out0[0] = mseE + mseF;
    }
  }
}

// ---------------------------------------------------------------------------
// Host entry
// ---------------------------------------------------------------------------
extern "C" void kernel_launch(void* const* d_in, const int* in_sizes, int n_in,
                              void* d_out, int out_size, void* d_ws, size_t ws_size,
                              hipStream_t stream)
{
  (void)in_sizes; (void)n_in; (void)out_size; (void)ws_size;
  const float* layout = (const float*)d_in[0];   // (1,2,NY,NX)
  const float* heat   = (const float*)d_in[1];   // (1,1,NY,NX)
  const float* flow   = (const float*)d_in[2];   // (1,2,NY,NX)
  float* out = (float*)d_out;                    // [scalar | heat_bc | eq_mask]
  float* partials = (float*)d_ws;                // 2 floats per block

  const size_t planeN = (size_t)NXg * NYg;
  dim3 grid(NXg / TX, NYg / TY);                 // 64 x 128 = 8192 blocks

  energy_stencil_kernel<<<grid, NTHREADS, 0, stream>>>(
      layout, heat, flow, out + 1, out + 1 + planeN, partials);
  energy_reduce_kernel<<<1, NTHREADS, 0, stream>>>(
      partials, (int)(grid.x * grid.y), out);
}